// HebbianMap2d_41068477284655
// MI455X (gfx1250) — compile-verified
//
#include <hip/hip_runtime.h>

typedef float v2f __attribute__((ext_vector_type(2)));
typedef float v8f __attribute__((ext_vector_type(8)));

#define B_  16
#define C_  3
#define H_  32
#define W_  32
#define KH  5
#define KW  5
#define HO  28
#define WO  28
#define O_  96
#define D_  75          // C*KH*KW
#define DP  76          // K padded to multiple of 4
#define N_  (B_*HO*WO)  // 12544
#define TILE_ROWS 32
#define NBLK (N_/TILE_ROWS)  // 392
#define ETA_F 1e-3f

__global__ void hebb_zero_ws(float* p, int n) {
    int i = blockIdx.x * blockDim.x + threadIdx.x;
    if (i < n) p[i] = 0.f;
}

__global__ __launch_bounds__(128)
void hebb_main(const float* __restrict__ x, const float* __restrict__ w,
               float* __restrict__ y, float* __restrict__ dw) {
    __shared__ float w_s[O_ * DP];          // 29184 B
    __shared__ float xu_s[TILE_ROWS * DP];  //  9728 B
    __shared__ float s_s[TILE_ROWS * O_];   // 12288 B

    const int tid = threadIdx.x;
    const int rowBase = blockIdx.x * TILE_ROWS;

    // --- stage weights into LDS, zero-padded K ---
    for (int t = tid; t < O_ * DP; t += 128) {
        int o = t / DP, k = t % DP;
        w_s[t] = (k < D_) ? w[o * D_ + k] : 0.f;
    }
    // --- build unfolded x tile (im2col gather) ---
    for (int t = tid; t < TILE_ROWS * DP; t += 128) {
        int r = t / DP, k = t % DP;
        float v = 0.f;
        if (k < D_) {
            int gn  = rowBase + r;
            int b   = gn / (HO * WO);
            int rem = gn % (HO * WO);
            int oh  = rem / WO, ow = rem % WO;
            int c    = k / (KH * KW);
            int rem2 = k % (KH * KW);
            int ki   = rem2 / KW, kj = rem2 % KW;
            v = x[((b * C_ + c) * H_ + (oh + ki)) * W_ + (ow + kj)];
        }
        xu_s[t] = v;
    }
    __syncthreads();

    // --- WMMA GEMM: s[32 x 96] = xu[32 x 76] * W^T[76 x 96] ---
    const int wave    = tid >> 5;       // 0..3
    const int lane    = tid & 31;
    const int rt      = wave >> 1;      // row tile 0..1
    const int ch      = wave & 1;       // column half (48 cols each)
    const int laneMod = lane & 15;
    const int hi      = lane >> 4;      // lane half
    const int koff    = hi * 2;         // K offset per ISA f32 A/B layout
    const int aRow    = rt * 16 + laneMod;
    const int cb0     = ch * 48;

    v8f acc0 = {}, acc1 = {}, acc2 = {};
    for (int kk = 0; kk < DP; kk += 4) {
        v2f a, b0, b1, b2;
        a.x  = xu_s[aRow * DP + kk + koff];
        a.y  = xu_s[aRow * DP + kk + koff + 1];
        b0.x = w_s[(cb0      + laneMod) * DP + kk + koff];
        b0.y = w_s[(cb0      + laneMod) * DP + kk + koff + 1];
        b1.x = w_s[(cb0 + 16 + laneMod) * DP + kk + koff];
        b1.y = w_s[(cb0 + 16 + laneMod) * DP + kk + koff + 1];
        b2.x = w_s[(cb0 + 32 + laneMod) * DP + kk + koff];
        b2.y = w_s[(cb0 + 32 + laneMod) * DP + kk + koff + 1];
        acc0 = __builtin_amdgcn_wmma_f32_16x16x4_f32(false, a, false, b0, (short)0, acc0, false, false);
        acc1 = __builtin_amdgcn_wmma_f32_16x16x4_f32(false, a, false, b1, (short)0, acc1, false, false);
        acc2 = __builtin_amdgcn_wmma_f32_16x16x4_f32(false, a, false, b2, (short)0, acc2, false, false);
    }

    // --- scatter C tiles: LDS s-tile + y_out ---
    #pragma unroll
    for (int v = 0; v < 8; ++v) {
        int r   = rt * 16 + v + hi * 8;     // C/D layout: M = v (+8 for hi lanes)
        int gn  = rowBase + r;
        int b   = gn / (HO * WO);
        int rem = gn % (HO * WO);
        int oh  = rem / WO, ow = rem % WO;
        int c0  = cb0 + laneMod;
        s_s[r * O_ + c0     ] = acc0[v];
        s_s[r * O_ + c0 + 16] = acc1[v];
        s_s[r * O_ + c0 + 32] = acc2[v];
        y[((b * O_ + c0     ) * HO + oh) * WO + ow] = acc0[v];
        y[((b * O_ + c0 + 16) * HO + oh) * WO + ow] = acc1[v];
        y[((b * O_ + c0 + 32) * HO + oh) * WO + ow] = acc2[v];
    }
    __syncthreads();

    // --- winner-take-all Hebbian delta accumulation (8 rows per wave) ---
    for (int r = wave * 8; r < wave * 8 + 8; ++r) {
        float v0 = s_s[r * O_ + lane];
        float v1 = s_s[r * O_ + lane + 32];
        float v2 = s_s[r * O_ + lane + 64];
        float m  = fmaxf(v0, fmaxf(v1, v2));
        #pragma unroll
        for (int off = 16; off > 0; off >>= 1)
            m = fmaxf(m, __shfl_xor(m, off, 32));

        unsigned masks[3];
        masks[0] = (unsigned)__ballot(v0 == m);
        masks[1] = (unsigned)__ballot(v1 == m);
        masks[2] = (unsigned)__ballot(v2 == m);

        const int d0 = lane, d1 = lane + 32, d2 = lane + 64;
        float pref0 = 0.f, pref1 = 0.f, pref2 = 0.f;
        // iterate winners in ascending column order (cumsum semantics, ties ok)
        for (int ww = 0; ww < 3; ++ww) {
            unsigned mm = masks[ww];
            while (mm) {
                int bit = __builtin_ctz(mm);
                mm &= mm - 1;
                int o = ww * 32 + bit;
                pref0 += m * w_s[o * DP + d0];
                pref1 += m * w_s[o * DP + d1];
                atomicAdd(&dw[o * D_ + d0], m * (xu_s[r * DP + d0] - pref0));
                atomicAdd(&dw[o * D_ + d1], m * (xu_s[r * DP + d1] - pref1));
                if (d2 < D_) {
                    pref2 += m * w_s[o * DP + d2];
                    atomicAdd(&dw[o * D_ + d2], m * (xu_s[r * DP + d2] - pref2));
                }
            }
        }
    }
}

__global__ void hebb_finalize(const float* __restrict__ w, const float* __restrict__ dw,
                              float* __restrict__ out, int n) {
    int i = blockIdx.x * blockDim.x + threadIdx.x;
    if (i < n) out[i] = w[i] + ETA_F * (dw[i] * (1.0f / (float)N_));
}

extern "C" void kernel_launch(void* const* d_in, const int* in_sizes, int n_in,
                              void* d_out, int out_size, void* d_ws, size_t ws_size,
                              hipStream_t stream) {
    const float* x = (const float*)d_in[0];   // [16,3,32,32]
    const float* w = (const float*)d_in[1];   // [96,3,5,5]
    float* y  = (float*)d_out;                // [16,96,28,28] = 1204224 floats
    float* nw = y + (size_t)B_ * O_ * HO * WO;// [96,3,5,5]    = 7200 floats
    float* dw = (float*)d_ws;                 // 7200-float delta accumulator

    hebb_zero_ws<<<(O_ * D_ + 255) / 256, 256, 0, stream>>>(dw, O_ * D_);
    hebb_main<<<NBLK, 128, 0, stream>>>(x, w, y, dw);
    hebb_finalize<<<(O_ * D_ + 255) / 256, 256, 0, stream>>>(w, dw, nw, O_ * D_);
}